// GRUModule_11596411699631
// MI455X (gfx1250) — compile-verified
//
#include <hip/hip_runtime.h>

// Problem constants from the reference: B=524288, A=4, C=9.
#define ROWS_TOTAL (524288 * 4)   // 2,097,152 independent GRU-cell rows
#define CDIM 9                    // [h, x0..x7]
#define TPB 256                   // 8 wave32 per block
#define ROWS_PER_BLOCK 1024       // 4 rows per thread
#define NBLOCKS (ROWS_TOTAL / ROWS_PER_BLOCK)          // 2048
#define TILE_FLOATS (ROWS_PER_BLOCK * CDIM)            // 9216 floats = 36864 B
#define TILE_F4 (TILE_FLOATS / 4)                      // 2304 float4
#define F4_PER_THREAD (TILE_F4 / TPB)                  // 9

// ---- gfx1250 async global->LDS path (guarded; falls back to plain staging) ----
#if defined(__has_builtin)
#if __has_builtin(__builtin_amdgcn_global_load_async_to_lds_b128) && \
    __has_builtin(__builtin_amdgcn_s_wait_asynccnt)
#define USE_ASYNC_LDS 1
#endif
#if __has_builtin(__builtin_amdgcn_tanhf)
#define HAVE_HW_TANH 1
#endif
#endif
#ifndef USE_ASYNC_LDS
#define USE_ASYNC_LDS 0
#endif
#ifndef HAVE_HW_TANH
#define HAVE_HW_TANH 0
#endif

// Exact pointer types the async builtin expects (from the clang diagnostic):
// arg0: int4 in addrspace(1) (printed "__device__"), arg1: int4 in addrspace(3).
typedef __attribute__((__vector_size__(4 * sizeof(int)))) int v4i_t;
typedef __attribute__((address_space(1))) v4i_t* gbl_v4i_p;
typedef __attribute__((address_space(3))) v4i_t* lds_v4i_p;

__device__ __forceinline__ float fast_sigmoid(float x) {
  return 1.0f / (1.0f + __expf(-x));
}

__device__ __forceinline__ float my_tanh(float x) {
#if HAVE_HW_TANH
  return __builtin_amdgcn_tanhf(x);   // v_tanh_f32 (gfx1250 TRANS unit)
#else
  return tanhf(x);
#endif
}

__global__ __launch_bounds__(TPB) void gru_cell_fused(
    const float* __restrict__ inputs,  // [ROWS_TOTAL, 9]
    const float* __restrict__ w_ih,    // [3, 8]
    const float* __restrict__ w_hh,    // [3, 1]
    const float* __restrict__ b_ih,    // [3]
    const float* __restrict__ b_hh,    // [3]
    const float* __restrict__ w_out,   // [1, 1]
    const float* __restrict__ b_out,   // [1]
    float* __restrict__ out)           // [ROWS_TOTAL]
{
  __shared__ __align__(16) float tile[TILE_FLOATS];

  const int t = threadIdx.x;
  const long long rowBase = (long long)blockIdx.x * ROWS_PER_BLOCK;

  // ---- Stage the 36 KB tile into LDS with coalesced 128-bit transfers ----
  const float4* __restrict__ gsrc = (const float4*)(inputs + rowBase * CDIM);
  float4* lds4 = (float4*)tile;

#if USE_ASYNC_LDS
  #pragma unroll
  for (int i = 0; i < F4_PER_THREAD; ++i) {
    const int idx = i * TPB + t;
    __builtin_amdgcn_global_load_async_to_lds_b128(
        (gbl_v4i_p)(gsrc + idx), (lds_v4i_p)(lds4 + idx),
        /*offset=*/0, /*cpol=*/0);
  }
  __builtin_amdgcn_s_wait_asynccnt(0);  // this wave's async copies done
#else
  #pragma unroll
  for (int i = 0; i < F4_PER_THREAD; ++i) {
    const int idx = i * TPB + t;
    lds4[idx] = gsrc[idx];
  }
#endif
  __syncthreads();  // all waves' tiles visible in LDS

  // ---- Tiny uniform weights -> scalar loads / SGPRs ----
  float W0[8], W1[8], W2[8];
  #pragma unroll
  for (int k = 0; k < 8; ++k) {
    W0[k] = w_ih[0 * 8 + k];
    W1[k] = w_ih[1 * 8 + k];
    W2[k] = w_ih[2 * 8 + k];
  }
  const float whh0 = w_hh[0], whh1 = w_hh[1], whh2 = w_hh[2];
  const float bih0 = b_ih[0], bih1 = b_ih[1], bih2 = b_ih[2];
  const float bhh0 = b_hh[0], bhh1 = b_hh[1], bhh2 = b_hh[2];
  const float wo = w_out[0], bo = b_out[0];

  // ---- Compute: rows t, t+256, t+512, t+768 (LDS stride 9 dwords, gcd(9,64)=1
  //      => bank-conflict-free across the wave; output stores coalesced) ----
  #pragma unroll
  for (int j = 0; j < ROWS_PER_BLOCK / TPB; ++j) {
    const int r = j * TPB + t;                   // row within tile
    const float* __restrict__ row = &tile[r * CDIM];

    const float h = row[0];
    float gi0 = bih0, gi1 = bih1, gi2 = bih2;
    #pragma unroll
    for (int k = 0; k < 8; ++k) {
      const float xv = row[1 + k];
      gi0 = fmaf(xv, W0[k], gi0);
      gi1 = fmaf(xv, W1[k], gi1);
      gi2 = fmaf(xv, W2[k], gi2);
    }
    const float gh0 = fmaf(h, whh0, bhh0);
    const float gh1 = fmaf(h, whh1, bhh1);
    const float gh2 = fmaf(h, whh2, bhh2);

    const float rg = fast_sigmoid(gi0 + gh0);
    const float zg = fast_sigmoid(gi1 + gh1);
    const float ng = my_tanh(fmaf(rg, gh2, gi2));
    const float hn = (1.0f - zg) * ng + zg * h;

    out[rowBase + r] = fmaf(hn, wo, bo);
  }
}

extern "C" void kernel_launch(void* const* d_in, const int* in_sizes, int n_in,
                              void* d_out, int out_size, void* d_ws, size_t ws_size,
                              hipStream_t stream) {
  (void)in_sizes; (void)n_in; (void)d_ws; (void)ws_size; (void)out_size;
  const float* inputs = (const float*)d_in[0];
  const float* w_ih   = (const float*)d_in[1];
  const float* w_hh   = (const float*)d_in[2];
  const float* b_ih   = (const float*)d_in[3];
  const float* b_hh   = (const float*)d_in[4];
  const float* w_out  = (const float*)d_in[5];
  const float* b_out  = (const float*)d_in[6];
  float* out = (float*)d_out;

  gru_cell_fused<<<NBLOCKS, TPB, 0, stream>>>(inputs, w_ih, w_hh, b_ih, b_hh,
                                              w_out, b_out, out);
}